// CopyRnnTF_51393578664139
// MI455X (gfx1250) — compile-verified
//
#include <hip/hip_runtime.h>

// ---------------------------------------------------------------------------
// CDNA5 (gfx1250) implementation of the pointer-generator copy-RNN reference.
// All matmuls use V_WMMA_F32_16X16X4_F32 (fp32 WMMA, wave32).
// ---------------------------------------------------------------------------

typedef __attribute__((ext_vector_type(2))) float v2f;
typedef __attribute__((ext_vector_type(8))) float v8f;

#define DEVINL __device__ __forceinline__

// Problem constants (match reference)
constexpr int kV   = 50000;
constexpr int kOOV = 100;
constexpr int kVO  = kV + kOOV;    // 50100
constexpr int kE   = 128;
constexpr int kH   = 256;
constexpr int kH4  = 4 * kH;       // 1024
constexpr int kB   = 16;
constexpr int kS   = 128;
constexpr int kT   = 128;

// fp32 WMMA: D(16x16) = A(16x4) * B(4x16) + C
// ISA layout: A: lane<16 -> row=lane, holds K=k+0,k+1; lane>=16 -> row=lane-16, K=k+2,k+3
//             B: lane<16 -> col=lane, holds K=k+0,k+1; lane>=16 -> col=lane-16, K=k+2,k+3
//             C/D: vgpr r, lane l -> row = 8*(l>>4)+r, col = l&15
DEVINL v8f wmma4(v2f a, v2f b, v8f c) {
  return __builtin_amdgcn_wmma_f32_16x16x4_f32(
      /*neg_a=*/false, a, /*neg_b=*/false, b,
      /*c_mod=*/(short)0, c, /*reuse_a=*/false, /*reuse_b=*/false);
}

DEVINL float sigmoidf_(float x) { return 1.0f / (1.0f + __expf(-x)); }

// ---------------------------------------------------------------------------
// Embedding gather: out[tok, e] = emb[ids[tok], e]   (E = 128)
// ---------------------------------------------------------------------------
__global__ void k_gather(const float* __restrict__ emb, const int* __restrict__ ids,
                         float* __restrict__ out, int n_tok) {
  int idx = blockIdx.x * blockDim.x + threadIdx.x;
  if (idx >= n_tok * kE) return;
  int tok = idx >> 7;           // /128
  int e   = idx & 127;
  out[idx] = emb[(long long)ids[tok] * kE + e];
}

// ---------------------------------------------------------------------------
// Generic fp32 WMMA GEMM: C[M,N] = act(A[M,K] @ B[K,N] + bias)
// One wave computes one 16x16 tile. ACT: 0=none, 1=tanh, 2=exp
// ---------------------------------------------------------------------------
template <int ACT>
__global__ void k_gemm(const float* __restrict__ A, const float* __restrict__ Bm,
                       const float* __restrict__ bias, float* __restrict__ C,
                       int M, int N, int K, int lda, int ldb, int ldc) {
  const int lane = threadIdx.x & 31;
  const int wid  = threadIdx.x >> 5;
  const int tilesN = N >> 4;
  const int tilesM = M >> 4;
  const long long tile = (long long)blockIdx.x * (blockDim.x >> 5) + wid;
  if (tile >= (long long)tilesM * tilesN) return;   // wave-uniform exit
  const int tm = (int)(tile / tilesN);
  const int tn = (int)(tile % tilesN);
  const int m0 = tm << 4, n0 = tn << 4;
  const int r  = lane & 15;
  const int kk = (lane >> 4) << 1;

  v8f acc = {};
  const float* Arow = A + (long long)(m0 + r) * lda;
  for (int k = 0; k < K; k += 4) {
    v2f a, b;
    a.x = Arow[k + kk];
    a.y = Arow[k + kk + 1];
    b.x = Bm[(long long)(k + kk)     * ldb + n0 + r];
    b.y = Bm[(long long)(k + kk + 1) * ldb + n0 + r];
    acc = wmma4(a, b, acc);
  }
  const int rowh = (lane >> 4) << 3;
  const int col  = n0 + r;
  const float bv = bias ? bias[col] : 0.0f;
#pragma unroll
  for (int i = 0; i < 8; ++i) {
    float v = acc[i] + bv;
    if (ACT == 1) v = tanhf(v);
    else if (ACT == 2) v = expf(v);
    C[(long long)(m0 + rowh + i) * ldc + col] = v;
  }
}

// ---------------------------------------------------------------------------
// Batched fp32 WMMA GEMM (per-batch strides). BT: B is [N,K] row-major (B^T).
// ---------------------------------------------------------------------------
template <int ACT, bool BT>
__global__ void k_gemm_b(const float* __restrict__ A, const float* __restrict__ Bm,
                         float* __restrict__ C,
                         int M, int N, int K, int lda, int ldb, int ldc,
                         long long sA, long long sB, long long sC) {
  const int bb = blockIdx.y;
  const float* Ab = A + (long long)bb * sA;
  const float* Bb = Bm + (long long)bb * sB;
  float* Cb = C + (long long)bb * sC;

  const int lane = threadIdx.x & 31;
  const int wid  = threadIdx.x >> 5;
  const int tilesN = N >> 4;
  const int tilesM = M >> 4;
  const long long tile = (long long)blockIdx.x * (blockDim.x >> 5) + wid;
  if (tile >= (long long)tilesM * tilesN) return;
  const int tm = (int)(tile / tilesN);
  const int tn = (int)(tile % tilesN);
  const int m0 = tm << 4, n0 = tn << 4;
  const int r  = lane & 15;
  const int kk = (lane >> 4) << 1;

  v8f acc = {};
  const float* Arow = Ab + (long long)(m0 + r) * lda;
  for (int k = 0; k < K; k += 4) {
    v2f a, b;
    a.x = Arow[k + kk];
    a.y = Arow[k + kk + 1];
    if (BT) {
      const float* Bn = Bb + (long long)(n0 + r) * ldb;
      b.x = Bn[k + kk];
      b.y = Bn[k + kk + 1];
    } else {
      b.x = Bb[(long long)(k + kk)     * ldb + n0 + r];
      b.y = Bb[(long long)(k + kk + 1) * ldb + n0 + r];
    }
    acc = wmma4(a, b, acc);
  }
  const int rowh = (lane >> 4) << 3;
  const int col  = n0 + r;
#pragma unroll
  for (int i = 0; i < 8; ++i) {
    float v = acc[i];
    if (ACT == 1) v = tanhf(v);
    else if (ACT == 2) v = expf(v);
    Cb[(long long)(m0 + rowh + i) * ldc + col] = v;
  }
}

// ---------------------------------------------------------------------------
// Persistent LSTM: one workgroup (512 threads = 16 waves), h/c in LDS.
// zpre: [B, L, 4H] = x@W + b (gate order i,f,g,o). U: [H, 4H].
// Wave w owns output columns [16w, 16w+16) of H, computing all four gate tiles
// (i: col, f: col+256, g: col+512, o: col+768) so the gate nonlinearity is
// applied directly on the WMMA accumulators.
// ---------------------------------------------------------------------------
__global__ void __launch_bounds__(512)
k_lstm(const float* __restrict__ zpre, const float* __restrict__ U,
       const int* __restrict__ x_len,        // nullptr -> no mask
       const float* __restrict__ h0,         // nullptr -> zeros
       const float* __restrict__ c0,
       float* __restrict__ hs,               // [B, L, H]
       float* __restrict__ hT, float* __restrict__ cT,
       int L) {
  __shared__ float sH[kB * kH];   // 16 KB
  __shared__ float sC[kB * kH];   // 16 KB

  const int tid  = threadIdx.x;
  const int lane = tid & 31;
  const int wid  = tid >> 5;      // 0..15

  for (int i = tid; i < kB * kH; i += 512) {
    sH[i] = h0 ? h0[i] : 0.0f;
    sC[i] = c0 ? c0[i] : 0.0f;
  }
  __syncthreads();

  const int r    = lane & 15;
  const int kk   = (lane >> 4) << 1;
  const int rowh = (lane >> 4) << 3;
  const int col  = (wid << 4) + r;     // column in [0, 256)

  for (int t = 0; t < L; ++t) {
    v8f zi = {}, zf = {}, zg = {}, zo = {};
    // seed accumulators with precomputed x@W + b
#pragma unroll
    for (int i = 0; i < 8; ++i) {
      const long long base = ((long long)(rowh + i) * L + t) * kH4;
      zi[i] = zpre[base + col];
      zf[i] = zpre[base + 256 + col];
      zg[i] = zpre[base + 512 + col];
      zo[i] = zpre[base + 768 + col];
    }
    // z += h @ U  (A from LDS, B from global/L2)
    for (int k = 0; k < kH; k += 4) {
      v2f a;
      a.x = sH[r * kH + k + kk];
      a.y = sH[r * kH + k + kk + 1];
      const float* U0 = U + (long long)(k + kk) * kH4 + col;
      const float* U1 = U0 + kH4;
      v2f b0, b1, b2, b3;
      b0.x = U0[0];   b0.y = U1[0];
      b1.x = U0[256]; b1.y = U1[256];
      b2.x = U0[512]; b2.y = U1[512];
      b3.x = U0[768]; b3.y = U1[768];
      zi = wmma4(a, b0, zi);
      zf = wmma4(a, b1, zf);
      zg = wmma4(a, b2, zg);
      zo = wmma4(a, b3, zo);
    }
    __syncthreads();   // all waves done reading sH for this step

#pragma unroll
    for (int i = 0; i < 8; ++i) {
      const int row = rowh + i;           // batch row 0..15
      const int idx = row * kH + col;
      const float ig = sigmoidf_(zi[i]);
      const float fg = sigmoidf_(zf[i]);
      const float og = sigmoidf_(zo[i]);
      const float gg = tanhf(zg[i]);
      const float cn = fg * sC[idx] + ig * gg;
      const float hn = og * tanhf(cn);
      const bool  m  = x_len ? (t < x_len[row]) : true;
      const float hf = m ? hn : sH[idx];
      const float cf = m ? cn : sC[idx];
      sH[idx] = hf;
      sC[idx] = cf;
      hs[((long long)row * L + t) * kH + col] = hf;
    }
    __syncthreads();   // writes visible before next step's reads
  }

  for (int i = tid; i < kB * kH; i += 512) {
    if (hT) hT[i] = sH[i];
    if (cT) cT[i] = sC[i];
  }
}

// ---------------------------------------------------------------------------
// Masked softmax over S per (b,t) row. Masked positions contribute score 0
// (exactly mirrors reference: softmax(mask * scores)). One wave per row.
// ---------------------------------------------------------------------------
__global__ void k_softmax(float* __restrict__ scores, const int* __restrict__ x_len) {
  const int lane = threadIdx.x & 31;
  const int row  = blockIdx.x * (blockDim.x >> 5) + (threadIdx.x >> 5);
  if (row >= kB * kT) return;
  const int b   = row / kT;
  const int len = x_len[b];
  float* p = scores + (long long)row * kS;

  float vals[4];
  float mx = -1e30f;
#pragma unroll
  for (int i = 0; i < 4; ++i) {
    const int s = lane + i * 32;
    const float v = (s < len) ? p[s] : 0.0f;
    vals[i] = v;
    mx = fmaxf(mx, v);
  }
  for (int off = 16; off; off >>= 1) mx = fmaxf(mx, __shfl_xor(mx, off, 32));
  float sum = 0.0f;
#pragma unroll
  for (int i = 0; i < 4; ++i) { vals[i] = expf(vals[i] - mx); sum += vals[i]; }
  for (int off = 16; off; off >>= 1) sum += __shfl_xor(sum, off, 32);
  const float inv = 1.0f / sum;
#pragma unroll
  for (int i = 0; i < 4; ++i) p[lane + i * 32] = vals[i] * inv;
}

// ---------------------------------------------------------------------------
// cat = [ctx | hs] along feature dim (2048 x 512)
// ---------------------------------------------------------------------------
__global__ void k_concat(const float* __restrict__ ctx, const float* __restrict__ hs,
                         float* __restrict__ cat) {
  const int idx = blockIdx.x * blockDim.x + threadIdx.x;
  if (idx >= kB * kT * 2 * kH) return;
  const int m = idx >> 9;          // /512
  const int k = idx & 511;
  cat[idx] = (k < kH) ? ctx[m * kH + k] : hs[m * kH + (k - kH)];
}

// summed[b,s] = sum_t exp(cs[b,t,s])
__global__ void k_sum_exp_t(const float* __restrict__ cs, float* __restrict__ summed) {
  const int idx = blockIdx.x * blockDim.x + threadIdx.x;
  if (idx >= kB * kS) return;
  const int b = idx / kS, s = idx % kS;
  const float* p = cs + (long long)b * kT * kS + s;
  float acc = 0.0f;
  for (int t = 0; t < kT; ++t) acc += expf(p[t * kS]);
  summed[idx] = acc;
}

// prob[b, t, V..V+OOV) = 1e-10 (the reference's pad)
__global__ void k_oov_fill(float* __restrict__ prob) {
  const int idx = blockIdx.x * blockDim.x + threadIdx.x;
  if (idx >= kB * kT * kOOV) return;
  const int bt = idx / kOOV;
  const int v  = idx % kOOV;
  prob[(long long)bt * kVO + kV + v] = 1e-10f;
}

// prob[b, t=s, x_with_oov[b,s]] += summed[b,s]  (unique (b,t) -> no atomics)
__global__ void k_scatter(float* __restrict__ prob, const int* __restrict__ xoov,
                          const float* __restrict__ summed) {
  const int idx = blockIdx.x * blockDim.x + threadIdx.x;
  if (idx >= kB * kS) return;
  const int b = idx / kS, s = idx % kS;
  const int v = xoov[idx];
  prob[((long long)(b * kT + s)) * kVO + v] += summed[idx];
}

// denom[b,v] = sum_t prob[b,t,v]
__global__ void k_denom(const float* __restrict__ prob, float* __restrict__ denom) {
  const long long idx = (long long)blockIdx.x * blockDim.x + threadIdx.x;
  if (idx >= (long long)kB * kVO) return;
  const int b = (int)(idx / kVO);
  const int v = (int)(idx % kVO);
  const float* p = prob + (long long)b * kT * kVO + v;
  float acc = 0.0f;
  for (int t = 0; t < kT; ++t) acc += p[(long long)t * kVO];
  denom[idx] = acc;
}

// prob = log1p(prob / denom[b,v])
__global__ void k_finalize(float* __restrict__ prob, const float* __restrict__ denom) {
  const long long idx = (long long)blockIdx.x * blockDim.x + threadIdx.x;
  if (idx >= (long long)kB * kT * kVO) return;
  const long long bt = idx / kVO;
  const int v = (int)(idx % kVO);
  const int b = (int)(bt / kT);
  prob[idx] = log1pf(prob[idx] / denom[(long long)b * kVO + v]);
}

// ---------------------------------------------------------------------------
// Host launcher
// ---------------------------------------------------------------------------
extern "C" void kernel_launch(void* const* d_in, const int* in_sizes, int n_in,
                              void* d_out, int out_size, void* d_ws, size_t ws_size,
                              hipStream_t stream) {
  (void)in_sizes; (void)n_in; (void)out_size; (void)ws_size;

  const int*   x       = (const int*)d_in[0];
  const int*   xoov    = (const int*)d_in[1];
  const int*   x_len   = (const int*)d_in[2];
  const int*   dec_x   = (const int*)d_in[3];
  const float* emb     = (const float*)d_in[4];
  const float* enc_W   = (const float*)d_in[5];
  const float* enc_U   = (const float*)d_in[6];
  const float* enc_b   = (const float*)d_in[7];
  const float* dec_W   = (const float*)d_in[8];
  const float* dec_U   = (const float*)d_in[9];
  const float* dec_b   = (const float*)d_in[10];
  const float* attn_W  = (const float*)d_in[11];
  const float* out_W   = (const float*)d_in[12];
  const float* out_b   = (const float*)d_in[13];
  const float* gen_W   = (const float*)d_in[14];
  const float* copy_W  = (const float*)d_in[15];
  const float* copy_b  = (const float*)d_in[16];

  float* out = (float*)d_out;
  // d_out layout: log1p(prob)[B,T,VO] | enc_out[B,S,H] | sh[B,H] | sc[B,H]
  const long long OUT_ENC = (long long)kB * kT * kVO;
  const long long OUT_SH  = OUT_ENC + (long long)kB * kS * kH;
  const long long OUT_SC  = OUT_SH + (long long)kB * kH;
  float* prob    = out;
  float* enc_out = out + OUT_ENC;
  float* sh_out  = out + OUT_SH;
  float* sc_out  = out + OUT_SC;

  // Workspace layout (floats)
  float* ws = (float*)d_ws;
  float* emb_x   = ws;                       // 262144
  float* emb_dec = emb_x   + 262144;         // 262144
  float* xw_enc  = emb_dec + 262144;         // 2097152
  float* xw_dec  = xw_enc  + 2097152;        // 2097152
  float* dec_hs  = xw_dec  + 2097152;        // 524288
  float* eh      = dec_hs  + 524288;         // 4096
  float* ec      = eh      + 4096;           // 4096
  float* hsW     = ec      + 4096;           // 524288
  float* scores  = hsW     + 524288;         // 262144 (aw in-place)
  float* ctx     = scores  + 262144;         // 524288
  float* catb    = ctx     + 524288;         // 1048576
  float* attn    = catb    + 1048576;        // 524288
  float* tenc    = attn    + 524288;         // 524288
  float* cs      = tenc    + 524288;         // 262144
  float* summed  = cs      + 262144;         // 2048
  float* denom   = summed  + 2048;           // 801600

  const int MT = kB * kT;   // 2048 rows (also B*S)
  auto cdivl = [](long long a, long long b) { return (int)((a + b - 1) / b); };

  // 1) Embedding gathers
  k_gather<<<cdivl((long long)MT * kE, 256), 256, 0, stream>>>(emb, x, emb_x, kB * kS);
  k_gather<<<cdivl((long long)MT * kE, 256), 256, 0, stream>>>(emb, dec_x, emb_dec, kB * kT);

  // 2) Pre-activation x@W + b for both LSTMs: [2048,128]@[128,1024]
  {
    long long tiles = (long long)(MT / 16) * (kH4 / 16);   // 8192
    k_gemm<0><<<cdivl(tiles, 8), 256, 0, stream>>>(emb_x, enc_W, enc_b, xw_enc,
                                                   MT, kH4, kE, kE, kH4, kH4);
    k_gemm<0><<<cdivl(tiles, 8), 256, 0, stream>>>(emb_dec, dec_W, dec_b, xw_dec,
                                                   MT, kH4, kE, kE, kH4, kH4);
  }

  // 3) Encoder LSTM (masked) -> enc_out (directly into d_out), eh, ec
  k_lstm<<<1, 512, 0, stream>>>(xw_enc, enc_U, x_len, nullptr, nullptr,
                                enc_out, eh, ec, kS);

  // 4) Decoder LSTM (unmasked, init from encoder state) -> dec_hs, sh, sc
  k_lstm<<<1, 512, 0, stream>>>(xw_dec, dec_U, nullptr, eh, ec,
                                dec_hs, sh_out, sc_out, kT);

  // 5) hsW = dec_hs @ attn_W : [2048,256]@[256,256]
  {
    long long tiles = (long long)(MT / 16) * (kH / 16);    // 2048
    k_gemm<0><<<cdivl(tiles, 8), 256, 0, stream>>>(dec_hs, attn_W, nullptr, hsW,
                                                   MT, kH, kH, kH, kH, kH);
  }

  // 6) scores[b,t,s] = hsW[b,t,:] . enc_out[b,s,:]  (batched, B^T)
  {
    dim3 g(cdivl((long long)(kT / 16) * (kS / 16), 8), kB);
    k_gemm_b<0, true><<<g, 256, 0, stream>>>(hsW, enc_out, scores,
                                             kT, kS, kH, kH, kH, kS,
                                             (long long)kT * kH, (long long)kS * kH,
                                             (long long)kT * kS);
  }

  // 7) masked softmax over S -> aw (in-place)
  k_softmax<<<cdivl(MT, 8), 256, 0, stream>>>(scores, x_len);

  // 8) ctx[b,t,h] = sum_s aw[b,t,s] * enc_out[b,s,h]
  {
    dim3 g(cdivl((long long)(kT / 16) * (kH / 16), 8), kB);
    k_gemm_b<0, false><<<g, 256, 0, stream>>>(scores, enc_out, ctx,
                                              kT, kH, kS, kS, kH, kH,
                                              (long long)kT * kS, (long long)kS * kH,
                                              (long long)kT * kH);
  }

  // 9) attn_out = tanh([ctx|hs] @ out_W + out_b) : [2048,512]@[512,256]
  k_concat<<<cdivl((long long)MT * 2 * kH, 256), 256, 0, stream>>>(ctx, dec_hs, catb);
  {
    long long tiles = (long long)(MT / 16) * (kH / 16);
    k_gemm<1><<<cdivl(tiles, 8), 256, 0, stream>>>(catb, out_W, out_b, attn,
                                                   MT, kH, 2 * kH, 2 * kH, kH, kH);
  }

  // 10) tenc = tanh(enc_out @ copy_W + copy_b) : [2048,256]@[256,256]
  {
    long long tiles = (long long)(MT / 16) * (kH / 16);
    k_gemm<1><<<cdivl(tiles, 8), 256, 0, stream>>>(enc_out, copy_W, copy_b, tenc,
                                                   MT, kH, kH, kH, kH, kH);
  }

  // 11) cs[b,t,s] = attn[b,t,:] . tenc[b,s,:] (batched, B^T); summed = sum_t exp(cs)
  {
    dim3 g(cdivl((long long)(kT / 16) * (kS / 16), 8), kB);
    k_gemm_b<0, true><<<g, 256, 0, stream>>>(attn, tenc, cs,
                                             kT, kS, kH, kH, kH, kS,
                                             (long long)kT * kH, (long long)kS * kH,
                                             (long long)kT * kS);
  }
  k_sum_exp_t<<<cdivl(kB * kS, 256), 256, 0, stream>>>(cs, summed);

  // 12) prob[:, :, 0:V] = exp(attn @ gen_W) : [2048,256]@[256,50000], ldc = VO
  {
    long long tiles = (long long)(MT / 16) * (kV / 16);    // 400000
    k_gemm<2><<<cdivl(tiles, 8), 256, 0, stream>>>(attn, gen_W, nullptr, prob,
                                                   MT, kV, kH, kH, kV, kVO);
  }

  // 13) OOV pad, copy-scatter, per-(b,v) sum over T, log1p(normalize)
  k_oov_fill<<<cdivl((long long)kB * kT * kOOV, 256), 256, 0, stream>>>(prob);
  k_scatter<<<cdivl(kB * kS, 256), 256, 0, stream>>>(prob, xoov, summed);
  k_denom<<<cdivl((long long)kB * kVO, 256), 256, 0, stream>>>(prob, denom);
  k_finalize<<<cdivl((long long)kB * kT * kVO, 256), 256, 0, stream>>>(prob, denom);
}